// SparseLinear2_44332652430011
// MI455X (gfx1250) — compile-verified
//
#include <hip/hip_runtime.h>

// SparseLinear2: out[b,g,v] = sum_{e: gene_ids[e]==g} x[b, feat_ids[e]] * w[e,v]
// gene_ids sorted -> per-gene contiguous edge segment -> tiny GEMM per gene,
// executed with V_WMMA_F32_16X16X4_F32 (full f32 precision).
// Memory-bound (~160 MB read / 20 MB write -> ~8 us HBM floor at 23.3 TB/s).
// This round: 32-bit element offsets everywhere (GVS + scale_offset addressing,
// no v_*_u64 address chains / xcnt stalls) and a fully scalar (SALU/SMEM)
// binary search via readfirstlane.

typedef __attribute__((ext_vector_type(2))) float v2f;
typedef __attribute__((ext_vector_type(8))) float v8f;

#define NUM_GENE_C 20000
#define BATCH_C 8
#define VDIM_C 32
#define WAVES_PER_BLOCK 8

// One K=4 WMMA step over edges (j0, j1) for this lane's K-pair.
// q0/q1 zero the A-matrix entries (covers both padded batch rows m>=8 and
// invalid tail K-slots). B is loaded unmasked from clamped (finite) addresses:
// a zero A column annihilates any finite B garbage.
__device__ __forceinline__ void chunk_mm(const float* __restrict__ x,
                                         const float* __restrict__ w,
                                         const int* __restrict__ feat_ids,
                                         int xrow, int m,
                                         int j0, int j1, bool q0, bool q1,
                                         v8f& acc_lo, v8f& acc_hi)
{
    // All offsets are 32-bit element counts -> global_load_b32 ... scale_offset
    const int f0  = feat_ids[j0];
    const int f1  = feat_ids[j1];
    const int w0o = j0 * VDIM_C + m;
    const int w1o = j1 * VDIM_C + m;

    // Unconditional loads (addresses always in range) ...
    const float xa0 = x[xrow + f0];
    const float xa1 = x[xrow + f1];
    const float b0l = w[w0o];
    const float b1l = w[w1o];
    const float b0h = w[w0o + 16];   // same voff as b0l, +64B immediate
    const float b1h = w[w1o + 16];

    // ... masking done on values -> v_cndmask, no EXEC branches.
    const float a0 = q0 ? xa0 : 0.0f;
    const float a1 = q1 ? xa1 : 0.0f;

    const v2f A  = {a0, a1};
    const v2f Bl = {b0l, b1l};
    const v2f Bh = {b0h, b1h};

    acc_lo = __builtin_amdgcn_wmma_f32_16x16x4_f32(
        false, A, false, Bl, (short)0, acc_lo, false, false);
    acc_hi = __builtin_amdgcn_wmma_f32_16x16x4_f32(
        false, A, false, Bh, (short)0, acc_hi, false, false);
}

__global__ __launch_bounds__(WAVES_PER_BLOCK * 32)
void sparse_linear2_wmma_kernel(const float* __restrict__ x,
                                const float* __restrict__ w,
                                const int* __restrict__ gene_ids,
                                const int* __restrict__ feat_ids,
                                float* __restrict__ out,
                                int nnz, int n_feat)
{
    // Pin the wave index into an SGPR so g/s/t and all loop control stay scalar.
    const int wv   = __builtin_amdgcn_readfirstlane((int)(threadIdx.x >> 5));
    const int g    = blockIdx.x * WAVES_PER_BLOCK + wv;   // scalar
    if (g >= NUM_GENE_C) return;                           // uniform

    const int lane = threadIdx.x & 31;
    const int m    = lane & 15;                // A row (batch) / B col (v)
    const int kb   = (lane >> 4) << 1;         // K-pair base: 0 or 2
    const bool am  = (m < BATCH_C);            // real batch row?
    // Padded rows (m>=8) gather the same x addresses as row m-8 -> coalesced,
    // then are zeroed by the 'am' mask.
    const int xrow = (m & 7) * n_feat;         // < 1.4M, fits int

    // Scalar binary searches over sorted gene_ids: [s, t) = edges of gene g.
    int s, t;
    {
        int lo = 0, hi = nnz;
        while (lo < hi) { int mid = (lo + hi) >> 1; if (gene_ids[mid] < g) lo = mid + 1; else hi = mid; }
        s = lo;
        hi = nnz;
        while (lo < hi) { int mid = (lo + hi) >> 1; if (gene_ids[mid] < g + 1) lo = mid + 1; else hi = mid; }
        t = lo;
    }

    v8f accA_lo = {0.f,0.f,0.f,0.f,0.f,0.f,0.f,0.f};
    v8f accA_hi = {0.f,0.f,0.f,0.f,0.f,0.f,0.f,0.f};
    v8f accB_lo = {0.f,0.f,0.f,0.f,0.f,0.f,0.f,0.f};
    v8f accB_hi = {0.f,0.f,0.f,0.f,0.f,0.f,0.f,0.f};

    int e0 = s;
    // Hot loop: 8 fully-valid edges per iteration -> zero predicates, two
    // independent accumulator sets so the WMMAs don't serialize.
    for (; e0 + 8 <= t; e0 += 8) {
        const int i0 = e0 + kb;
        chunk_mm(x, w, feat_ids, xrow, m, i0,     i0 + 1, am, am, accA_lo, accA_hi);
        chunk_mm(x, w, feat_ids, xrow, m, i0 + 4, i0 + 5, am, am, accB_lo, accB_hi);
    }
    // Tail: <=7 edges; clamp addresses into [s, t-1], mask A values.
    for (; e0 < t; e0 += 4) {
        const int i0  = e0 + kb;
        const int i1  = i0 + 1;
        const int tm1 = t - 1;                 // t > s here, so valid
        const int j0  = i0 < tm1 ? i0 : tm1;
        const int j1  = i1 < tm1 ? i1 : tm1;
        const bool q0 = am & (i0 < t);
        const bool q1 = am & (i1 < t);
        chunk_mm(x, w, feat_ids, xrow, m, j0, j1, q0, q1, accA_lo, accA_hi);
    }

    const v8f acc_lo = accA_lo + accB_lo;
    const v8f acc_hi = accA_hi + accB_hi;

    // D (16x16 f32): VGPR r, lanes 0-15 hold M=r (batch r), N=lane.
    // Rows M>=8 (lanes 16-31) are padding -> only lanes 0-15 store.
    if (lane < 16) {
        const int gbase = g * VDIM_C + lane;   // 32-bit offsets (max ~5.2M elems)
        #pragma unroll
        for (int b = 0; b < BATCH_C; ++b) {
            const int base = b * (NUM_GENE_C * VDIM_C) + gbase;
            out[base]      = acc_lo[b];
            out[base + 16] = acc_hi[b];
        }
    }
}

extern "C" void kernel_launch(void* const* d_in, const int* in_sizes, int n_in,
                              void* d_out, int out_size, void* d_ws, size_t ws_size,
                              hipStream_t stream) {
    const float* x        = (const float*)d_in[0];   // (8, N_FEAT) f32
    const float* w        = (const float*)d_in[1];   // (NNZ, 32)   f32
    const int*   gene_ids = (const int*)d_in[2];     // (NNZ,) sorted i32
    const int*   feat_ids = (const int*)d_in[3];     // (NNZ,) i32
    float*       out      = (float*)d_out;           // (8, 20000, 32) f32

    const int nnz    = in_sizes[2];
    const int n_feat = in_sizes[0] / BATCH_C;

    const int threads = WAVES_PER_BLOCK * 32;
    const int blocks  = (NUM_GENE_C + WAVES_PER_BLOCK - 1) / WAVES_PER_BLOCK;
    sparse_linear2_wmma_kernel<<<blocks, threads, 0, stream>>>(
        x, w, gene_ids, feat_ids, out, nnz, n_feat);
}